// EnhancedWaveletTransform2D_32134945308966
// MI455X (gfx1250) — compile-verified
//
#include <hip/hip_runtime.h>
#include <hip/hip_bf16.h>
#include <math.h>

// ---------------------------------------------------------------------------
// EnhancedWaveletTransform2D for MI455X (gfx1250, wave32)
//   x: [8,256,128,128] f32.  Wavelet round-trip == identity, so pipeline is:
//   pool -> SE-MLP (WMMA f32 16x16x4) -> scale -> dw3x3 conv -> InstanceNorm
//   -> LeakyReLU.  Memory-bound: floor ~256 MiB HBM traffic (~11us @ 23.3TB/s).
// ---------------------------------------------------------------------------

#define B_  8
#define C_  256
#define HW_ 128
#define PLANE (HW_ * HW_)          // 16384 elements = 64 KiB
#define SLOPE 0.01f
#define EPS_  1e-5f

typedef __attribute__((ext_vector_type(2))) float v2f;
typedef __attribute__((ext_vector_type(8))) float v8f;

// ======================= Kernel 1: global average pool ======================
// One 256-thread workgroup per (b,c) plane.  p[bc] = mean(x[bc])
__global__ __launch_bounds__(256) void pool_kernel(const float* __restrict__ x,
                                                   float* __restrict__ p) {
  __shared__ float red[256];
  const int t  = threadIdx.x;
  const int bc = blockIdx.x;
  const float4* xp = (const float4*)(x + (size_t)bc * PLANE);

  float s = 0.0f;
#pragma unroll
  for (int i = 0; i < 16; ++i) {
    float4 v = xp[i * 256 + t];
    s += (v.x + v.y) + (v.z + v.w);
  }
  red[t] = s;
  __syncthreads();
#pragma unroll
  for (int off = 128; off > 0; off >>= 1) {
    if (t < off) red[t] += red[t + off];
    __syncthreads();
  }
  if (t == 0) p[bc] = red[0] * (1.0f / (float)PLANE);
}

// =================== Kernel 2: SE attention via f32 WMMA ====================
// H(16x64) = leaky(P(16x256) @ W1^T),  A(16x256) = sigmoid(H @ W2^T)
// Uses V_WMMA_F32_16X16X4_F32 so the math stays exact f32 like the reference.
// P is staged (zero-padded to 16 rows) into LDS first so the WMMA inner loops
// are divergence-free: ds_load_b64 (A) + global_load_b64 (B) + v_wmma.
// Operand layouts (ISA 7.12.2, 32-bit, wave32):
//   A 16x4 : lane l -> m = l%16, half = l/16; vgpr i holds A[m][2*half+i]
//   B 4x16 : lane l -> n = l%16, half = l/16; vgpr i holds B[2*half+i][n]
//   C/D    : lane l -> n = l%16; vgpr r holds D[r + 8*(l/16)][n]
__global__ __launch_bounds__(512) void attn_gemm_kernel(
    const float* __restrict__ p,   // [8,256] pooled means
    const float* __restrict__ w1,  // [64,256]
    const float* __restrict__ w2,  // [256,64]
    float* __restrict__ a_out) {   // [8,256] sigmoid gates
  __shared__ float Ps[16 * C_];    // padded P, 16 KiB
  __shared__ float Hs[16 * 64];    // hidden tile, 4 KiB

  const int t    = threadIdx.x;
  const int lane = t & 31;
  const int wave = t >> 5;        // 0..15
  const int m    = lane & 15;     // tile row (GEMM M) / tile col for B
  const int half = lane >> 4;     // selects K pair {0,1} vs {2,3}

  // ---- stage P into LDS with rows 8..15 zeroed (uniform, divergence-free) --
#pragma unroll
  for (int idx = 0; idx < 16 * C_; idx += 512) {
    const int j   = idx + t;
    const int row = j >> 8;            // /256
    Ps[j] = (row < B_) ? p[j] : 0.0f;  // p is [8,256] contiguous
  }
  __syncthreads();

  // ---- GEMM1: waves 0..3 each compute one 16x16 N-tile of H ----
  if (wave < 4) {
    const int nt = wave;
    v8f acc = {};
    const float* arow = &Ps[m * C_];                 // A row (LDS)
    const float* brow = &w1[(nt * 16 + m) * C_];     // B column source (global)
    for (int kk = 0; kk < 64; ++kk) {                // K = 256 in steps of 4
      const int k0 = kk * 4 + half * 2;
      v2f av, bv;
      av.x = arow[k0];
      av.y = arow[k0 + 1];
      // B[k][n] = W1^T[k][nt*16+n] = w1[(nt*16+n)*256 + k]
      bv.x = brow[k0];
      bv.y = brow[k0 + 1];
      acc = __builtin_amdgcn_wmma_f32_16x16x4_f32(
          false, av, false, bv, (short)0, acc, false, false);
    }
#pragma unroll
    for (int r = 0; r < 8; ++r) {
      float v = acc[r];
      v = fmaxf(v, SLOPE * v);                       // LeakyReLU
      Hs[(r + 8 * half) * 64 + nt * 16 + m] = v;
    }
  }
  __syncthreads();

  // ---- GEMM2: wave w computes N-tile w of the [16 x 256] logits ----
  {
    const int nt = wave;                             // 0..15
    v8f acc = {};
    const float* arow = &Hs[m * 64];                 // A row (LDS)
    const float* brow = &w2[(nt * 16 + m) * 64];     // B column source (global)
    for (int kk = 0; kk < 16; ++kk) {                // K = 64 in steps of 4
      const int k0 = kk * 4 + half * 2;
      v2f av, bv;
      av.x = arow[k0];
      av.y = arow[k0 + 1];
      // B[k][n] = W2^T[k][nt*16+n] = w2[(nt*16+n)*64 + k]
      bv.x = brow[k0];
      bv.y = brow[k0 + 1];
      acc = __builtin_amdgcn_wmma_f32_16x16x4_f32(
          false, av, false, bv, (short)0, acc, false, false);
    }
    if (half == 0) {                                 // rows 0..7 == batches 0..7
#pragma unroll
      for (int r = 0; r < 8; ++r) {
        float v = acc[r];
        v = 1.0f / (1.0f + expf(-v));                // sigmoid
        a_out[r * C_ + nt * 16 + m] = v;
      }
    }
  }
}

// === Kernel 3: fused scale + depthwise 3x3 + InstanceNorm + LeakyReLU =======
// One 256-thread workgroup per (b,c) plane; whole plane staged in LDS with a
// zero halo (SAME padding).  Conv outputs live in registers while moments are
// accumulated, then a single normalize+store pass.  x read once (L2-resident),
// out written once -> HBM floor.
#define LDS_W 132                 // 128 + 2 halo, padded to dodge bank conflicts
#define LDS_H 130                 // 128 + 2 halo
__global__ __launch_bounds__(256) void fuse_kernel(
    const float* __restrict__ x, const float* __restrict__ a,
    const float* __restrict__ rw, const float* __restrict__ rb,
    float* __restrict__ out) {
  __shared__ float xs[LDS_H * LDS_W];          // ~68.6 KiB (<< 320 KB/WGP)
  __shared__ float redA[256];
  __shared__ float redB[256];
  __shared__ float stat[2];

  const int t  = threadIdx.x;
  const int bc = blockIdx.x;
  const int c  = bc & (C_ - 1);
  const float scale = a[bc];

  float w[9];
#pragma unroll
  for (int i = 0; i < 9; ++i) w[i] = rw[c * 9 + i];
  const float bias = rb[c];

  // zero the halo ring
  for (int i = t; i < LDS_W; i += 256) {
    xs[i] = 0.0f;
    xs[(LDS_H - 1) * LDS_W + i] = 0.0f;
  }
  for (int r = t; r < HW_; r += 256) {
    xs[(r + 1) * LDS_W + 0]   = 0.0f;
    xs[(r + 1) * LDS_W + 129] = 0.0f;
    xs[(r + 1) * LDS_W + 130] = 0.0f;
    xs[(r + 1) * LDS_W + 131] = 0.0f;
  }

  // stage scaled plane into LDS interior (coalesced float4 loads)
  const float4* xp = (const float4*)(x + (size_t)bc * PLANE);
#pragma unroll
  for (int i = 0; i < 16; ++i) {
    const int chunk = i * 256 + t;             // 0..4095, 4 px each
    float4 v = xp[chunk];
    const int pix = chunk << 2;
    const int r   = pix >> 7;                  // /128
    const int col = pix & 127;
    float* dst = &xs[(r + 1) * LDS_W + col + 1];
    dst[0] = v.x * scale; dst[1] = v.y * scale;
    dst[2] = v.z * scale; dst[3] = v.w * scale;
  }
  __syncthreads();

  // depthwise 3x3 conv; keep 64 outputs/thread in registers, accumulate moments
  float yv[16][4];
  float s = 0.0f, s2 = 0.0f;
#pragma unroll
  for (int i = 0; i < 16; ++i) {
    const int chunk = i * 256 + t;
    const int pix = chunk << 2;
    const int r   = pix >> 7;
    const int col = pix & 127;
    const float* base = &xs[r * LDS_W + col];  // top-left tap of first output
    float rowbuf[3][6];
#pragma unroll
    for (int dr = 0; dr < 3; ++dr)
#pragma unroll
      for (int dc = 0; dc < 6; ++dc) rowbuf[dr][dc] = base[dr * LDS_W + dc];
#pragma unroll
    for (int j = 0; j < 4; ++j) {
      float acc = bias;
#pragma unroll
      for (int dr = 0; dr < 3; ++dr)
#pragma unroll
        for (int dc = 0; dc < 3; ++dc)
          acc = fmaf(w[dr * 3 + dc], rowbuf[dr][j + dc], acc);
      yv[i][j] = acc;
      s  += acc;
      s2 += acc * acc;
    }
  }

  // block-reduce sum and sum-of-squares
  redA[t] = s;
  redB[t] = s2;
  __syncthreads();
#pragma unroll
  for (int off = 128; off > 0; off >>= 1) {
    if (t < off) { redA[t] += redA[t + off]; redB[t] += redB[t + off]; }
    __syncthreads();
  }
  if (t == 0) {
    const float mu  = redA[0] * (1.0f / (float)PLANE);
    const float var = redB[0] * (1.0f / (float)PLANE) - mu * mu;  // biased var
    stat[0] = mu;
    stat[1] = rsqrtf(var + EPS_);
  }
  __syncthreads();
  const float mu = stat[0], rstd = stat[1];

  // normalize + LeakyReLU + coalesced float4 store
  float4* op = (float4*)(out + (size_t)bc * PLANE);
#pragma unroll
  for (int i = 0; i < 16; ++i) {
    const float z0 = (yv[i][0] - mu) * rstd;
    const float z1 = (yv[i][1] - mu) * rstd;
    const float z2 = (yv[i][2] - mu) * rstd;
    const float z3 = (yv[i][3] - mu) * rstd;
    float4 v;
    v.x = fmaxf(z0, SLOPE * z0);
    v.y = fmaxf(z1, SLOPE * z1);
    v.z = fmaxf(z2, SLOPE * z2);
    v.w = fmaxf(z3, SLOPE * z3);
    op[i * 256 + t] = v;
  }
}

// ============================== launcher ====================================
extern "C" void kernel_launch(void* const* d_in, const int* in_sizes, int n_in,
                              void* d_out, int out_size, void* d_ws, size_t ws_size,
                              hipStream_t stream) {
  const float* x  = (const float*)d_in[0];   // [8,256,128,128]
  const float* w1 = (const float*)d_in[1];   // [64,256]
  const float* w2 = (const float*)d_in[2];   // [256,64]
  const float* rw = (const float*)d_in[3];   // [256,1,3,3]
  const float* rb = (const float*)d_in[4];   // [256]
  float* out = (float*)d_out;

  float* p = (float*)d_ws;                   // [8*256] pooled means
  float* a = p + B_ * C_;                    // [8*256] sigmoid gates

  pool_kernel<<<B_ * C_, 256, 0, stream>>>(x, p);
  attn_gemm_kernel<<<1, 512, 0, stream>>>(p, w1, w2, a);
  fuse_kernel<<<B_ * C_, 256, 0, stream>>>(x, a, rw, rb, out);
}